// GateRecurrent2dnoind_65180423685494
// MI455X (gfx1250) — compile-verified
//
#include <hip/hip_runtime.h>
#include <stdint.h>

// Problem constants from the reference: X,B,G1,G2,G3 all (8, 96, 128, 128) f32.
#define NDIM   8
#define CDIM   96
#define HDIM   128
#define WDIM   128
#define PLANES (NDIM * CDIM)      // 768 independent (H,W) planes
#define NARR   5                  // X, B, G1, G2, G3

#define WT     16                 // w-columns per LDS tile
#define NT     (WDIM / WT)        // 8 tiles
#define ST     20                 // padded tile row stride (floats): 80B rows -> 16B-aligned B128 DMA
#define TILEF  (HDIM * ST)        // floats per (array, buffer) tile = 2560
#define STAGEF (NARR * 2 * TILEF) // 25600 floats of staging (double buffered)
#define HCF    (HDIM + 2)         // recurrent column + zero halos
#define SMEM_FLOATS (STAGEF + 2 * HCF)
#define SMEM_BYTES  (SMEM_FLOATS * 4)   // ~101 KB dynamic LDS (< 320KB/WGP on CDNA5)

// Issue the async DMA of one 128x16 tile (5 arrays) into LDS buffer `buf`.
// Per array: 16 wave-wide global_load_async_to_lds_b128 instructions; each lane
// moves 16B. Lane mapping: 4 lanes cover one h-row's 16 w's (64B contiguous
// global bursts); LDS side lands at the padded-stride location (16B aligned).
__device__ __forceinline__ void async_tile_load(
    uint32_t smemBase, int buf, int w0, int wave, int lane,
    const float* __restrict__ s0, const float* __restrict__ s1,
    const float* __restrict__ s2, const float* __restrict__ s3,
    const float* __restrict__ s4)
{
#define ISSUE_ARR(A, SRC)                                                        \
  {                                                                              \
    uint32_t sb = smemBase + (uint32_t)(((A) * 2 + buf) * TILEF) * 4u;           \
    _Pragma("unroll")                                                            \
    for (int j = 0; j < 4; ++j) {                                                \
      int i  = wave + 4 * j;            /* 16 instrs per array, 4 per wave */    \
      int m  = i * 32 + lane;           /* 0..511 : which 16B chunk */           \
      int h  = m >> 2;                  /* row */                                \
      int c4 = (m & 3) << 2;            /* w sub-chunk (0,4,8,12) */             \
      uint64_t ga = (uint64_t)(uintptr_t)((SRC) + (size_t)h * WDIM + w0 + c4);   \
      uint32_t la = sb + (uint32_t)(h * ST + c4) * 4u;                           \
      asm volatile("global_load_async_to_lds_b128 %0, %1, off"                   \
                   :: "v"(la), "v"(ga) : "memory");                              \
    }                                                                            \
  }
  ISSUE_ARR(0, s0)
  ISSUE_ARR(1, s1)
  ISSUE_ARR(2, s2)
  ISSUE_ARR(3, s3)
  ISSUE_ARR(4, s4)
#undef ISSUE_ARR
}

__global__ __launch_bounds__(HDIM)
void GateRecurrent2d_kernel(const float* __restrict__ X,  const float* __restrict__ B,
                            const float* __restrict__ G1, const float* __restrict__ G2,
                            const float* __restrict__ G3, float* __restrict__ O)
{
  extern __shared__ float smem[];

  const int t    = threadIdx.x;        // h row, 0..127
  const int wave = t >> 5;
  const int lane = t & 31;
  const size_t pbase = (size_t)blockIdx.x * (HDIM * WDIM);

  const float* x  = X  + pbase;
  const float* b  = B  + pbase;
  const float* g1 = G1 + pbase;
  const float* g2 = G2 + pbase;
  const float* g3 = G3 + pbase;
  float*       o  = O  + pbase;

  // Low 32 bits of a generic LDS pointer == byte offset in the LDS aperture.
  const uint32_t smemBase = (uint32_t)(uintptr_t)smem;

  // Recurrent column ping-pong with zero halos at [0] and [HDIM+1].
  float* hcA = smem + STAGEF;
  float* hcB = hcA + HCF;
  hcA[t + 1] = 0.0f;                          // h_prev(w = -1) == 0
  hcB[t + 1] = 0.0f;
  if (t < 2) {                                // halos: indices 0 and HCF-1
    hcA[t * (HCF - 1)] = 0.0f;
    hcB[t * (HCF - 1)] = 0.0f;
  }

  // Prime the pipeline: DMA tile 0 into buffer 0.
  async_tile_load(smemBase, 0, 0, wave, lane, x, b, g1, g2, g3);
  asm volatile("s_wait_asynccnt 0" ::: "memory");
  __syncthreads();

  float* hp = hcA;   // previous column (read)
  float* hn = hcB;   // new column (write)
  int cur = 0;

  for (int tile = 0; tile < NT; ++tile) {
    // Overlap: DMA tile+1 into the other buffer while we scan this one.
    if (tile + 1 < NT)
      async_tile_load(smemBase, cur ^ 1, (tile + 1) * WT, wave, lane, x, b, g1, g2, g3);

    const int r = t * ST;
    const float* sx = smem + (0 * 2 + cur) * TILEF + r;
    const float* sB = smem + (1 * 2 + cur) * TILEF + r;
    const float* s1 = smem + (2 * 2 + cur) * TILEF + r;
    const float* s2 = smem + (3 * 2 + cur) * TILEF + r;
    const float* s3 = smem + (4 * 2 + cur) * TILEF + r;

    float4 acc;
#pragma unroll
    for (int s = 0; s < WT; ++s) {
      const float hu = hp[t];        // h_prev[h-1]  (halo -> 0 at h==0)
      const float hm = hp[t + 1];    // h_prev[h]
      const float hd = hp[t + 2];    // h_prev[h+1]  (halo -> 0 at h==127)
      const float hv = sB[s] * sx[s] + s1[s] * hu + s2[s] * hm + s3[s] * hd;
      hn[t + 1] = hv;

      if      ((s & 3) == 0) acc.x = hv;
      else if ((s & 3) == 1) acc.y = hv;
      else if ((s & 3) == 2) acc.z = hv;
      else                   acc.w = hv;
      if ((s & 3) == 3)      // 16B burst store every 4 columns
        *(float4*)(o + (size_t)t * WDIM + tile * WT + (s - 3)) = acc;

      __syncthreads();       // one barrier per step; ping-pong gives RAW/WAR safety
      float* tmp = hp; hp = hn; hn = tmp;
    }

    // Next tile's DMA (issued above) must have landed before we read it.
    asm volatile("s_wait_asynccnt 0" ::: "memory");
    __syncthreads();
    cur ^= 1;
  }
}

extern "C" void kernel_launch(void* const* d_in, const int* in_sizes, int n_in,
                              void* d_out, int out_size, void* d_ws, size_t ws_size,
                              hipStream_t stream) {
  (void)in_sizes; (void)n_in; (void)out_size; (void)d_ws; (void)ws_size;
  const float* X  = (const float*)d_in[0];
  const float* B  = (const float*)d_in[1];
  const float* G1 = (const float*)d_in[2];
  const float* G2 = (const float*)d_in[3];
  const float* G3 = (const float*)d_in[4];
  float* O = (float*)d_out;

  GateRecurrent2d_kernel<<<PLANES, HDIM, SMEM_BYTES, stream>>>(X, B, G1, G2, G3, O);
}